// FeatureSelection_34230889349593
// MI455X (gfx1250) — compile-verified
//
#include <hip/hip_runtime.h>
#include <math.h>

typedef __attribute__((ext_vector_type(2))) float v2f;
typedef __attribute__((ext_vector_type(4))) float v4f;
typedef __attribute__((ext_vector_type(8))) float v8f;

#define B_ROWS 16384
#define D_EMB  128
#define D_HID  1024
#define D_FEAT 4096

// ---------------------------------------------------------------------------
// Kernel 1: h[g] = relu(ctx[g] @ W1[g] + b1[g]);  g in {0,1}; h is [2, D_HID]
// 2048 threads total; column dot over K=128, loads coalesced across threads.
// ---------------------------------------------------------------------------
__global__ void gate_hidden_kernel(const float* __restrict__ ctx1,
                                   const float* __restrict__ ctx2,
                                   const float* __restrict__ W1a,
                                   const float* __restrict__ b1a,
                                   const float* __restrict__ W1b,
                                   const float* __restrict__ b1b,
                                   float* __restrict__ h) {
  int j  = blockIdx.x * blockDim.x + threadIdx.x;  // 0..2047
  int g  = j >> 10;
  int jj = j & (D_HID - 1);
  const float* ctx = g ? ctx2 : ctx1;
  const float* W1  = g ? W1b  : W1a;
  const float* b1  = g ? b1b  : b1a;
  float acc = 0.f;
#pragma unroll 8
  for (int k = 0; k < D_EMB; ++k)
    acc = fmaf(ctx[k], W1[k * D_HID + jj], acc);
  acc += b1[jj];
  h[g * D_HID + jj] = acc > 0.f ? acc : 0.f;
}

// ---------------------------------------------------------------------------
// Kernel 2: gt[g] = 2*sigmoid(h[g] @ W2[g] + b2[g])   via v_wmma_f32_16x16x4_f32
// One wave per 16-column output tile (512 tiles = 2 gates * 256 tiles).
// A-matrix (16x4 f32): all 16 rows replicate the h row -> every row of D is
// the desired dot; we read row 0 (D VGPR0, lanes 0..15). A and B use the same
// K-slot convention (VGPR0 = {K0 | K2}, VGPR1 = {K1 | K3} across lane halves),
// so the K reduction aligns slot-for-slot.
// ---------------------------------------------------------------------------
__global__ void gate_out_wmma_kernel(const float* __restrict__ h,
                                     const float* __restrict__ W2a,
                                     const float* __restrict__ b2a,
                                     const float* __restrict__ W2b,
                                     const float* __restrict__ b2b,
                                     float* __restrict__ gt) {
  const int lane = threadIdx.x & 31;
  const int wave = (blockIdx.x * blockDim.x + threadIdx.x) >> 5;  // 0..511
  const int g      = wave >> 8;                // gate select
  const int n_base = (wave & 255) * 16;        // output column tile
  const float* hh = h + g * D_HID;
  const float* W2 = g ? W2b : W2a;
  const float* b2 = g ? b2b : b2a;
  const int half = lane >> 4;   // 0: K-slots {0,1}; 1: K-slots {2,3}
  const int n    = lane & 15;

  v8f c = {};
  for (int k0 = 0; k0 < D_HID; k0 += 4) {
    const int ka = k0 + 2 * half;
    v2f a, b;
    a.x = hh[ka];
    a.y = hh[ka + 1];
    b.x = W2[(size_t)(ka)     * D_FEAT + n_base + n];
    b.y = W2[(size_t)(ka + 1) * D_FEAT + n_base + n];
    c = __builtin_amdgcn_wmma_f32_16x16x4_f32(
        /*neg_a=*/false, a, /*neg_b=*/false, b,
        /*c_mod=*/(short)0, c, /*reuse_a=*/false, /*reuse_b=*/false);
  }
  if (lane < 16) {
    const int j = n_base + n;           // D row 0: VGPR0, lanes 0..15, N=lane
    const float x = c[0] + b2[j];
    gt[g * D_FEAT + j] = 2.0f / (1.0f + __expf(-x));
  }
}

// ---------------------------------------------------------------------------
// Kernel 3: bandwidth workhorse.  out1 = emb * gt1 ; out2 = emb * gt2.
// Each thread owns one float4 column (gates held in registers), streams 32
// rows with non-temporal b128 loads/stores (touch-once data; keep L2 clean).
// grid = (D_FEAT/4/256, B/32) = (4, 512), block = 256 (8 wave32s).
// ---------------------------------------------------------------------------
__global__ void broadcast_mul_kernel(const float* __restrict__ emb,
                                     const float* __restrict__ gt,
                                     float* __restrict__ out) {
  const int col4 = blockIdx.x * 256 + threadIdx.x;        // 0..1023
  const v4f* emb4  = (const v4f*)emb;
  const v4f* gt1_4 = (const v4f*)gt;
  const v4f* gt2_4 = (const v4f*)(gt + D_FEAT);
  v4f* out1 = (v4f*)out;
  v4f* out2 = out1 + (size_t)B_ROWS * (D_FEAT / 4);

  const v4f g1 = gt1_4[col4];
  const v4f g2 = gt2_4[col4];

  const int r0 = blockIdx.y * 32;
  size_t idx = (size_t)r0 * (D_FEAT / 4) + col4;
#pragma unroll 4
  for (int r = 0; r < 32; ++r) {
    v4f e = __builtin_nontemporal_load(&emb4[idx]);
    __builtin_nontemporal_store(e * g1, &out1[idx]);
    __builtin_nontemporal_store(e * g2, &out2[idx]);
    idx += (D_FEAT / 4);
  }
}

// ---------------------------------------------------------------------------
extern "C" void kernel_launch(void* const* d_in, const int* in_sizes, int n_in,
                              void* d_out, int out_size, void* d_ws, size_t ws_size,
                              hipStream_t stream) {
  const float* flat_emb = (const float*)d_in[0];
  const float* c1       = (const float*)d_in[1];
  const float* c2       = (const float*)d_in[2];
  const float* fs1_W1   = (const float*)d_in[3];
  const float* fs1_b1   = (const float*)d_in[4];
  const float* fs1_W2   = (const float*)d_in[5];
  const float* fs1_b2   = (const float*)d_in[6];
  const float* fs2_W1   = (const float*)d_in[7];
  const float* fs2_b1   = (const float*)d_in[8];
  const float* fs2_W2   = (const float*)d_in[9];
  const float* fs2_b2   = (const float*)d_in[10];

  float* out = (float*)d_out;
  float* h   = (float*)d_ws;          // [2, D_HID]
  float* gt  = h + 2 * D_HID;         // [2, D_FEAT]

  // 1) hidden layer of both gates (2048 outputs)
  gate_hidden_kernel<<<dim3(8), dim3(256), 0, stream>>>(
      c1, c2, fs1_W1, fs1_b1, fs2_W1, fs2_b1, h);

  // 2) output layer of both gates via WMMA (512 waves, 8 waves/block)
  gate_out_wmma_kernel<<<dim3(64), dim3(256), 0, stream>>>(
      h, fs1_W2, fs1_b2, fs2_W2, fs2_b2, gt);

  // 3) streaming broadcast multiply: 768 MB of traffic, HBM-bound
  broadcast_mul_kernel<<<dim3(4, 512), dim3(256), 0, stream>>>(
      flat_emb, gt, out);
}